// DDCModel_26448408609317
// MI455X (gfx1250) — compile-verified
//
#include <hip/hip_runtime.h>

// Problem constants (match reference).
#define S 4096        // state-space size
#define A 4           // actions
#define B 8           // trajectories
#define L 128         // steps
#define KCHUNK 1024   // K staging chunk (9*KCHUNK*4 = 36 KB LDS incl. zero pad row)
#define NBLK (S / 64) // GEMM n-tile blocks per action

typedef float v8f __attribute__((ext_vector_type(8)));
typedef float v2f __attribute__((ext_vector_type(2)));
typedef unsigned int u32x4 __attribute__((ext_vector_type(4)));
typedef int i32x4 __attribute__((ext_vector_type(4)));
typedef int i32x8 __attribute__((ext_vector_type(8)));

#if defined(__HIP_DEVICE_COMPILE__)
#if !__has_builtin(__builtin_amdgcn_wmma_f32_16x16x4_f32)
#error "missing __builtin_amdgcn_wmma_f32_16x16x4_f32 on this toolchain (device pass)"
#endif
#if __has_builtin(__builtin_amdgcn_tensor_load_to_lds) && \
    __has_builtin(__builtin_amdgcn_s_wait_tensorcnt)
#define HAVE_TDM 1
#else
#define HAVE_TDM 0
#endif
#else
#define HAVE_TDM 0
#endif

// Build one-hot interface0 in buf0.
__global__ __launch_bounds__(256) void init_iface(const int* __restrict__ init_states,
                                                  float* __restrict__ iface) {
    for (int i = threadIdx.x; i < B * S; i += blockDim.x) iface[i] = 0.0f;
    __syncthreads();
    if (threadIdx.x < B) iface[threadIdx.x * S + init_states[threadIdx.x]] = 1.0f;
}

// One step. Grid: (NBLK + 1, A), block = 128 threads (4 waves).
//  blockIdx.x <  NBLK : dst[b, ntile] = src[b, :] @ T[a][:, ntile]  (row b written
//                       iff actions[b,t] == a  -> fused one_hot select, no all_new buffer)
//  blockIdx.x == NBLK : reward[b,t] = dot(src[b,:], R[a,:]) for rows with action a.
__global__ __launch_bounds__(128) void step_kernel(const float* __restrict__ src,
                                                   float* __restrict__ dst,
                                                   const float* __restrict__ Tm,
                                                   const float* __restrict__ Rm,
                                                   const int* __restrict__ actions,
                                                   int t, float* __restrict__ out) {
    __shared__ __align__(16) float sA[(B + 1) * KCHUNK];  // row 8 = zero padding

    const int a = blockIdx.y;

    // Which trajectories took action `a` at step t?
    int match = 0;
    #pragma unroll
    for (int r = 0; r < B; ++r) match |= (actions[r * L + t] == a) ? (1 << r) : 0;
    if (match == 0) return;  // no row uses T[a] this step: skip its 64 MB stream

    // ---- Reward blocks -------------------------------------------------
    if (blockIdx.x == NBLK) {
        float* red = sA;  // reuse LDS for the block reduction
        const float* __restrict__ Rv = Rm + (size_t)a * S;
        for (int r = 0; r < B; ++r) {
            if (!(match & (1 << r))) continue;  // uniform per block
            float p = 0.0f;
            for (int k = threadIdx.x; k < S; k += 128)
                p += src[(size_t)r * S + k] * Rv[k];
            red[threadIdx.x] = p;
            __syncthreads();
            #pragma unroll
            for (int sft = 64; sft > 0; sft >>= 1) {
                if (threadIdx.x < sft) red[threadIdx.x] += red[threadIdx.x + sft];
                __syncthreads();
            }
            if (threadIdx.x == 0) out[r * L + t] = red[0];
            __syncthreads();
        }
        return;
    }

    // ---- GEMM blocks ---------------------------------------------------
    const int wave = threadIdx.x >> 5;
    const int lane = threadIdx.x & 31;
    const int half = lane >> 4;        // A-frag: 0 -> K{0,1}, 1 -> K{2,3}
    const int row  = lane & 15;        // A-frag M index / B-frag N index
    const int n0   = (blockIdx.x * 4 + wave) * 16;

    const float* __restrict__ Ta = Tm + ((size_t)a << 24);  // a * S * S
    // Rows 8..15 of the padded A matrix read the zeroed LDS pad row (branchless).
    const float* ap = &sA[((row < B) ? row : B) * KCHUNK];

    // Zero the pad row once (visible after the first in-loop barrier).
    for (int i = threadIdx.x; i < KCHUNK; i += 128) sA[B * KCHUNK + i] = 0.0f;

    v8f c = {};  // f32 16x16 accumulator (8 VGPRs)

    for (int kbase = 0; kbase < S; kbase += KCHUNK) {
        __syncthreads();  // previous chunk fully consumed (and pad row visible)

#if HAVE_TDM
        // Stage src[0:8][kbase : kbase+KCHUNK] -> sA via the Tensor Data Mover.
        if (threadIdx.x < 32) {  // one wave issues the DMA
            const unsigned long long ga =
                (unsigned long long)(uintptr_t)(src + kbase);
            const unsigned int lds = (unsigned int)(uintptr_t)&sA[0];  // addr[31:0]
            u32x4 g0;
            g0[0] = 1u;                                   // count=1, user-mode D#
            g0[1] = lds;                                  // lds_addr
            g0[2] = (unsigned int)(ga & 0xFFFFFFFFu);     // global_addr[31:0]
            g0[3] = (unsigned int)((ga >> 32) & 0x01FFFFFFu) | (2u << 30); // type=2
            i32x8 g1;
            g1[0] = (int)(2u << 16);       // data_size=4B, wg_mask=0, no flags
            g1[1] = (int)(4096u << 16);    // tensor_dim0[15:0]=4096 @ bits[63:48]
            g1[2] = (int)((unsigned)B << 16); // tensor_dim0 hi=0; tensor_dim1 lo=8
            g1[3] = (int)(1024u << 16);    // tensor_dim1 hi=0; tile_dim0=1024
            g1[4] = B;                     // tile_dim1=8, tile_dim2=0
            g1[5] = 4096;                  // tensor_dim0_stride[31:0]
            g1[6] = 0;                     // stride0 hi, tensor_dim1_stride lo
            g1[7] = 0;
            const i32x4 gz = {0, 0, 0, 0};
#if __clang_major__ >= 23
            const i32x8 gz8 = {0, 0, 0, 0, 0, 0, 0, 0};
            __builtin_amdgcn_tensor_load_to_lds(g0, g1, gz, gz, gz8, 0);
#else
            __builtin_amdgcn_tensor_load_to_lds(g0, g1, gz, gz, 0);
#endif
            __builtin_amdgcn_s_wait_tensorcnt(0);
        }
#else
        // Fallback: coalesced b128 copy loop.
        for (int i = threadIdx.x; i < (B * KCHUNK) / 4; i += 128) {
            const int rr = (i * 4) / KCHUNK;
            const int cc = (i * 4) % KCHUNK;
            *(float4*)(&sA[rr * KCHUNK + cc]) =
                *(const float4*)(src + (size_t)rr * S + kbase + cc);
        }
#endif
        __syncthreads();

        // 2-stage software pipeline: issue frags for k+4, WMMA on k.
        const float* bp = Ta + (size_t)(kbase + 2 * half) * S + n0 + row;
        v2f av_n = *(const v2f*)(ap + 2 * half);
        v2f bv_n;
        bv_n.x = bp[0];
        bv_n.y = bp[S];

        #pragma unroll 4
        for (int k = 0; k < KCHUNK - 4; k += 4) {
            const v2f av = av_n;
            const v2f bv = bv_n;
            bp += (size_t)4 * S;
            av_n = *(const v2f*)(ap + (k + 4) + 2 * half);
            bv_n.x = bp[0];
            bv_n.y = bp[S];
            __builtin_prefetch(bp + (size_t)16 * S, 0, 1);  // global_prefetch_b8
            c = __builtin_amdgcn_wmma_f32_16x16x4_f32(
                    false, av, false, bv, (short)0, c, false, false);
        }
        c = __builtin_amdgcn_wmma_f32_16x16x4_f32(
                false, av_n, false, bv_n, (short)0, c, false, false);
    }

    // D layout: lanes 0-15 hold rows 0..7 in c[0..7], N = n0 + lane.
    if (lane < 16) {
        const int col = n0 + lane;
        #pragma unroll
        for (int r = 0; r < B; ++r) {
            if (match & (1 << r)) dst[(size_t)r * S + col] = c[r];
        }
    }
}

extern "C" void kernel_launch(void* const* d_in, const int* in_sizes, int n_in,
                              void* d_out, int out_size, void* d_ws, size_t ws_size,
                              hipStream_t stream) {
    const int*   init_states = (const int*)d_in[0];   // [B]
    const int*   actions     = (const int*)d_in[1];   // [B, L]
    const float* Tm          = (const float*)d_in[2]; // [A, S, S]
    const float* R           = (const float*)d_in[3]; // [A, S]
    float*       out         = (float*)d_out;         // [B, L]

    float* buf0 = (float*)d_ws;   // [B, S] ping
    float* buf1 = buf0 + B * S;   // [B, S] pong  (256 KB of d_ws total)

    init_iface<<<1, 256, 0, stream>>>(init_states, buf0);

    dim3 grid(NBLK + 1, A);
    for (int t = 0; t < L; ++t) {
        float* src = (t & 1) ? buf1 : buf0;
        float* dst = (t & 1) ? buf0 : buf1;
        step_kernel<<<grid, 128, 0, stream>>>(src, dst, Tm, R, actions, t, out);
    }
}